// MultifreqFstar_18700287607198
// MI455X (gfx1250) — compile-verified
//
#include <hip/hip_runtime.h>

// CDNA5 / gfx1250, wave32. f32 WMMA 16x16x4 throughout.

typedef __attribute__((ext_vector_type(2))) float v2f;
typedef __attribute__((ext_vector_type(8))) float v8f;

#define NXC 64
#define NKC 16
#define BATCHC 16
#define FEATC (NXC * NXC)

// ---------------------------------------------------------------------------
// Stage 1: per (f, batch, c): gather D_R, D_I (64x64), then for 4 terms
//   out[k] += sum_i A_t[i,k] * (D @ B_t)[i,k]
// WMMA f32 16x16x4 matmuls (i-tiles processed in pairs to reuse B-frags)
// + per-lane weighted row reduction using the documented C/D layout.
// Writes polar[fb=f*16+batch][q=c*64+k] into workspace.
// ---------------------------------------------------------------------------
__global__ __launch_bounds__(128) void polar_stage_kernel(
    const float* __restrict__ inputs,   // (16, 2, 4096, 16)
    const int*   __restrict__ r_index,  // (262144,)
    const float* __restrict__ pre1,  const float* __restrict__ pre2,
    const float* __restrict__ pre3,  const float* __restrict__ pre4,
    const float* __restrict__ post1, const float* __restrict__ post2,
    const float* __restrict__ post3, const float* __restrict__ post4,
    const float* __restrict__ ck1, const float* __restrict__ sk1,
    const float* __restrict__ ck2, const float* __restrict__ sk2,
    const float* __restrict__ ck3, const float* __restrict__ sk3,
    const float* __restrict__ ck4, const float* __restrict__ sk4,
    float* __restrict__ polar)          // (256, 4096) workspace
{
    __shared__ float sD[2][NXC * NXC];  // gathered R, I tiles (32 KB)
    __shared__ float sA[NXC * NXC];     // current term A = post (.) kernel2 (16 KB)
    __shared__ float sB[NXC * NXC];     // current term B = pre  (.) kernel1 (16 KB)

    const int tid  = threadIdx.x;
    const int lane = tid & 31;
    const int wave = tid >> 5;          // 4 waves; wave == k-tile index

    const int c     = blockIdx.x;       // 0..63
    const int batch = blockIdx.y;       // 0..15
    const int f     = blockIdx.z;       // 0..15

    // ---- gather D_R, D_I from inputs through r_index ----
    const int* rix = r_index + c * FEATC;
    const float* baseR = inputs + ((size_t)(batch * 2 + 0) * FEATC) * NKC + f;
    const float* baseI = inputs + ((size_t)(batch * 2 + 1) * FEATC) * NKC + f;
    for (int e = tid; e < FEATC; e += 128) {
        const int idx = rix[e];
        sD[0][e] = baseR[(size_t)idx * NKC];
        sD[1][e] = baseI[(size_t)idx * NKC];
    }
    __syncthreads();

    // term tables: A[i,k] = post[i]*K2[i,k] ; B[j,k] = pre[j]*K1[j,k]
    const float* PRE[4]  = {pre1,  pre2,  pre3,  pre4};
    const float* POST[4] = {post1, post2, post3, post4};
    const float* K2[4]   = {ck1, sk1, ck3, sk4};   // A-side kernels
    const float* K1[4]   = {ck2, sk2, sk3, ck4};   // B-side kernels
    const int    DSEL[4] = {0, 0, 1, 1};           // R, R, I, I

    const int kcol    = wave * 16 + (lane & 15);   // my output column k
    const int halfsel = lane >> 4;                 // 0: lanes 0-15, 1: 16-31
    const int arow    = (lane & 15);               // A-frag row within i-tile
    float acc = 0.0f;                              // partial sum_i for column k

    for (int t = 0; t < 4; ++t) {
        __syncthreads();   // previous term done reading sA/sB
        const float* k2f   = K2[t] + f * FEATC;
        const float* k1f   = K1[t] + f * FEATC;
        const float* postf = POST[t] + f * NXC;
        const float* pref  = PRE[t]  + f * NXC;
        for (int e = tid; e < FEATC; e += 128) {
            const int row = e >> 6;
            sA[e] = postf[row] * k2f[e];
            sB[e] = pref[row]  * k1f[e];
        }
        __syncthreads();

        const float* Dt = sD[DSEL[t]];

        for (int it = 0; it < 4; it += 2) {
            v8f m0 = {};
            v8f m1 = {};
            #pragma unroll
            for (int j = 0; j < NXC; j += 4) {
                // B-frag shared by both i-tiles: rows j..j+3, N col = kcol
                v2f b;
                b.x = sB[(j + halfsel * 2)     * NXC + kcol];
                b.y = sB[(j + halfsel * 2 + 1) * NXC + kcol];
                // A-frags: D rows for the two i-tiles, K cols j..j+3
                const float* a0p = &Dt[((it    ) * 16 + arow) * NXC + j + halfsel * 2];
                const float* a1p = &Dt[((it + 1) * 16 + arow) * NXC + j + halfsel * 2];
                v2f a0; a0.x = a0p[0]; a0.y = a0p[1];
                v2f a1; a1.x = a1p[0]; a1.y = a1p[1];
                m0 = __builtin_amdgcn_wmma_f32_16x16x4_f32(
                         false, a0, false, b, (short)0, m0, false, false);
                m1 = __builtin_amdgcn_wmma_f32_16x16x4_f32(
                         false, a1, false, b, (short)0, m1, false, false);
            }
            // reduce: lanes 0-15 hold rows r, lanes 16-31 rows r+8 of each tile
            const int rb0 = (it    ) * 16 + halfsel * 8;
            const int rb1 = (it + 1) * 16 + halfsel * 8;
            #pragma unroll
            for (int r = 0; r < 8; ++r) {
                acc += sA[(rb0 + r) * NXC + kcol] * m0[r];
                acc += sA[(rb1 + r) * NXC + kcol] * m1[r];
            }
        }
    }

    // combine the two half-wave partials for the same column k
    acc += __shfl_xor(acc, 16, 32);
    if (lane < 16) {
        const size_t fb = (size_t)f * BATCHC + batch;
        polar[fb * FEATC + c * NXC + wave * 16 + lane] = acc;
    }
}

// ---------------------------------------------------------------------------
// Stage 2: out_cart[fb, p] = sum_q polar[fb, q] * cart[p, q]
// GEMM M=256, N=4096, K=4096 via WMMA f32 16x16x4. Each block: 64(M)x128(N),
// LDS-tiled K chunks of 64, 2x2 result tiles per wave (A/B frag reuse),
// next-chunk prefetch (global_prefetch_b8), scatter epilogue into
// (batch, eta1, eta2, f) layout.
// ---------------------------------------------------------------------------
__global__ __launch_bounds__(256) void cart_stage_kernel(
    const float* __restrict__ cart,    // (4096, 4096) [p][q]
    const float* __restrict__ polar,   // (256, 4096)  [fb][q]
    float* __restrict__ out)           // (16, 64, 64, 16) [batch][e1][e2][f]
{
    __shared__ float sP[64 * 64];      // polar chunk [m][qq]   (16 KB)
    __shared__ float sC[64 * 128];     // cart^T chunk [qq][p'] (32 KB)

    const int tid  = threadIdx.x;
    const int lane = tid & 31;
    const int wave = tid >> 5;         // 8 waves

    const int n0 = blockIdx.x * 128;   // p base   (32 blocks)
    const int m0 = blockIdx.y * 64;    // fb base  (4 blocks)

    const int mt0 = (wave >> 2) * 2;   // M-tile pair base: 0 or 2
    const int nt0 = (wave & 3) * 2;    // N-tile pair base: 0,2,4,6

    const int halfsel = lane >> 4;
    const int ml      = lane & 15;

    v8f acc00 = {};
    v8f acc01 = {};
    v8f acc10 = {};
    v8f acc11 = {};

    for (int q0 = 0; q0 < 4096; q0 += 64) {
        __syncthreads();
        for (int e = tid; e < 4096; e += 256) {
            const int m = e >> 6, qq = e & 63;
            sP[e] = polar[(size_t)(m0 + m) * 4096 + q0 + qq];
        }
        // transpose cart tile while staging: sC[qq][p] = cart[n0+p][q0+qq]
        for (int e = tid; e < 8192; e += 256) {
            const int p = e >> 6, qq = e & 63;
            sC[qq * 128 + p] = cart[(size_t)(n0 + p) * 4096 + q0 + qq];
        }
        // prefetch next K-chunk of cart into cache while we do WMMA
        if (q0 + 64 < 4096 && tid < 128) {
            __builtin_prefetch(&cart[(size_t)(n0 + tid) * 4096 + q0 + 64], 0, 1);
        }
        __syncthreads();

        const int n_a = (nt0    ) * 16 + ml;
        const int n_b = (nt0 + 1) * 16 + ml;
        #pragma unroll
        for (int j = 0; j < 64; j += 4) {
            const float* a0p = &sP[((mt0    ) * 16 + ml) * 64 + j + halfsel * 2];
            const float* a1p = &sP[((mt0 + 1) * 16 + ml) * 64 + j + halfsel * 2];
            v2f a0; a0.x = a0p[0]; a0.y = a0p[1];
            v2f a1; a1.x = a1p[0]; a1.y = a1p[1];
            v2f b0, b1;
            b0.x = sC[(j + halfsel * 2)     * 128 + n_a];
            b0.y = sC[(j + halfsel * 2 + 1) * 128 + n_a];
            b1.x = sC[(j + halfsel * 2)     * 128 + n_b];
            b1.y = sC[(j + halfsel * 2 + 1) * 128 + n_b];
            acc00 = __builtin_amdgcn_wmma_f32_16x16x4_f32(
                        false, a0, false, b0, (short)0, acc00, false, false);
            acc01 = __builtin_amdgcn_wmma_f32_16x16x4_f32(
                        false, a0, false, b1, (short)0, acc01, false, false);
            acc10 = __builtin_amdgcn_wmma_f32_16x16x4_f32(
                        false, a1, false, b0, (short)0, acc10, false, false);
            acc11 = __builtin_amdgcn_wmma_f32_16x16x4_f32(
                        false, a1, false, b1, (short)0, acc11, false, false);
        }
    }

    // epilogue: scatter to out[batch][e1][e2][f], f = fb>>4, batch = fb&15
    #pragma unroll
    for (int mi = 0; mi < 2; ++mi) {
        #pragma unroll
        for (int ni = 0; ni < 2; ++ni) {
            const v8f acc = (mi == 0) ? (ni == 0 ? acc00 : acc01)
                                      : (ni == 0 ? acc10 : acc11);
            const int p  = n0 + (nt0 + ni) * 16 + ml;
            const int e1 = p >> 6, e2 = p & 63;
            #pragma unroll
            for (int r = 0; r < 8; ++r) {
                const int fb = m0 + (mt0 + mi) * 16 + halfsel * 8 + r;
                const int fo = fb >> 4, bo = fb & 15;
                out[(((size_t)bo * 64 + e1) * 64 + e2) * NKC + fo] = acc[r];
            }
        }
    }
}

extern "C" void kernel_launch(void* const* d_in, const int* in_sizes, int n_in,
                              void* d_out, int out_size, void* d_ws, size_t ws_size,
                              hipStream_t stream) {
    const float* inputs  = (const float*)d_in[0];
    const float* cart    = (const float*)d_in[1];
    const int*   r_index = (const int*)  d_in[2];
    const float* pre1  = (const float*)d_in[3];
    const float* pre2  = (const float*)d_in[4];
    const float* pre3  = (const float*)d_in[5];
    const float* pre4  = (const float*)d_in[6];
    const float* post1 = (const float*)d_in[7];
    const float* post2 = (const float*)d_in[8];
    const float* post3 = (const float*)d_in[9];
    const float* post4 = (const float*)d_in[10];
    const float* ck1 = (const float*)d_in[11];
    const float* sk1 = (const float*)d_in[12];
    const float* ck2 = (const float*)d_in[13];
    const float* sk2 = (const float*)d_in[14];
    const float* ck3 = (const float*)d_in[15];
    const float* sk3 = (const float*)d_in[16];
    const float* ck4 = (const float*)d_in[17];
    const float* sk4 = (const float*)d_in[18];

    float* polar = (float*)d_ws;   // 256 * 4096 * 4B = 4 MB scratch

    polar_stage_kernel<<<dim3(64, 16, 16), 128, 0, stream>>>(
        inputs, r_index,
        pre1, pre2, pre3, pre4, post1, post2, post3, post4,
        ck1, sk1, ck2, sk2, ck3, sk3, ck4, sk4, polar);

    cart_stage_kernel<<<dim3(32, 4), 256, 0, stream>>>(
        cart, polar, (float*)d_out);
}